// Quantizer_83751862272679
// MI455X (gfx1250) — compile-verified
//
#include <hip/hip_runtime.h>

typedef __attribute__((ext_vector_type(2))) float v2f;
typedef __attribute__((ext_vector_type(8))) float v8f;
typedef int v4i32 __attribute__((vector_size(16)));   // builtin's expected pointee

#define M_TOTAL 16384
#define K_CODES 8192
#define CDIM    256
#define HWSZ    1024          // H*W = 32*32
#define MROWS   32            // M rows per block (2 WMMA M-tiles)
#define NBLOCKS (M_TOTAL / MROWS)   // 512
#define BPITCH  36            // padded row pitch (floats) for B chunk in LDS

#if defined(__gfx1250__) && __has_builtin(__builtin_amdgcn_global_load_async_to_lds_b128) && __has_builtin(__builtin_amdgcn_s_wait_asynccnt)
#define USE_ASYNC 1
#else
#define USE_ASYNC 0
#endif

// ---------------------------------------------------------------------------
// Kernel 1: per-code squared norms ||e_k||^2  (one wave32 per code row)
// ---------------------------------------------------------------------------
__global__ __launch_bounds__(256) void vq_enorm(const float* __restrict__ e,
                                                float* __restrict__ enorm) {
    int wave = (blockIdx.x * blockDim.x + threadIdx.x) >> 5;
    int lane = threadIdx.x & 31;
    if (wave >= K_CODES) return;
    const float* row = e + (size_t)wave * CDIM;
    float s = 0.f;
    #pragma unroll
    for (int c = 0; c < CDIM; c += 32) {
        float v = row[c + lane];
        s += v * v;
    }
    #pragma unroll
    for (int m = 16; m >= 1; m >>= 1) s += __shfl_xor(s, m, 32);
    if (lane == 0) enorm[wave] = s;
}

// ---------------------------------------------------------------------------
// Kernel 2: main quantizer. One block per 32-row M slab (2 M-tiles, 8 waves).
//  - z slab staged in LDS as two conflict-free [c][16] planes (32 KB)
//  - codebook streamed in 256-code x 32-channel chunks, double-buffered in LDS
//    via async global->LDS b128 copies (s_wait_asynccnt + barrier to publish)
//  - wave w owns codes [sg*256 + w*32, +32): 2 K-tiles x 2 M-tiles = 4 accums
//  - fp32 WMMA 16x16x4 accumulation; argmin of ||e||^2 - 2*dot, ties->low idx
// ---------------------------------------------------------------------------
__global__ __launch_bounds__(256) void vq_main(const float* __restrict__ z,
                                               const float* __restrict__ e,
                                               const float* __restrict__ enorm,
                                               float* __restrict__ z_out,
                                               float* __restrict__ idx_out,
                                               float* __restrict__ loss_partial) {
    __shared__ float zT[2][CDIM][16];        // 32 KB, A tiles (pitch 16: conflict-free)
    __shared__ float bB[2][256][BPITCH];     // 72 KB, double-buffered B chunks
    __shared__ float wBest[8][MROWS];
    __shared__ int   wIdx[8][MROWS];
    __shared__ int   finalIdx[MROWS];
    __shared__ float lred[8];

    const int tid  = threadIdx.x;
    const int wave = tid >> 5;
    const int lane = tid & 31;
    const int half = lane >> 4;    // K-pair select for A/B frags; M-row-half for C/D
    const int nl   = lane & 15;

    const int m0 = blockIdx.x * MROWS;
    const int n  = m0 >> 10;                    // batch index (slab = one full h row)
    const int hw = m0 & (HWSZ - 1);             // h*32, w spans 0..31
    const long long zbase = (long long)n * CDIM * HWSZ + hw;   // + c*HWSZ + mi

    // ---- stage z slab: zT[mt][c][mi&15] = z[n, c, h, mi] ----
    #pragma unroll
    for (int j = 0; j < MROWS; ++j) {
        int li = j * 256 + tid;
        int c  = li >> 5;
        int mi = li & 31;
        zT[mi >> 4][c][mi & 15] = z[zbase + (long long)c * HWSZ + mi];
    }

    // ---- async staging of one 256-code x 32-channel chunk (32 KB) ----
    auto stage = [&](int sg, int cc, int buf) {
        const float* src = e + (size_t)sg * 256 * CDIM + cc * 32;
        #pragma unroll
        for (int r = 0; r < 8; ++r) {
            int li   = r * 256 + tid;     // one 16B quad per step
            int code = li >> 3;
            int q4   = (li & 7) * 4;
#if USE_ASYNC
            __builtin_amdgcn_global_load_async_to_lds_b128(
                (v4i32*)(src + (size_t)code * CDIM + q4),
                (v4i32*)(&bB[buf][code][q4]), 0, 0);
#else
            float4 v = *(const float4*)(src + (size_t)code * CDIM + q4);
            *(float4*)(&bB[buf][code][q4]) = v;
#endif
        }
    };

    float best[2][8];
    int   bidx[2][8];
    #pragma unroll
    for (int mt = 0; mt < 2; ++mt)
        #pragma unroll
        for (int r = 0; r < 8; ++r) { best[mt][r] = 3.402823e38f; bidx[mt][r] = 0; }

    stage(0, 0, 0);
    int buf = 0;

    for (int sg = 0; sg < 32; ++sg) {           // 256 codes per super-group
        const int kbase = sg * 256;
        v8f acc[2][2];                           // [mtile][ktile]
        acc[0][0] = (v8f){}; acc[0][1] = (v8f){};
        acc[1][0] = (v8f){}; acc[1][1] = (v8f){};

        for (int cc = 0; cc < 8; ++cc) {        // 32-channel chunks
            // issue next chunk into the other buffer
            int nsg = sg, ncc = cc + 1;
            if (ncc == 8) { ncc = 0; ++nsg; }
            if (nsg < 32) {
                stage(nsg, ncc, buf ^ 1);
#if USE_ASYNC
                __builtin_amdgcn_s_wait_asynccnt(8);   // prev chunk's 8 ops done
#endif
            } else {
#if USE_ASYNC
                __builtin_amdgcn_s_wait_asynccnt(0);
#endif
            }
            __syncthreads();                    // publish chunk `buf` to all waves

            const int c0b = cc * 32;
            #pragma unroll
            for (int s = 0; s < 8; ++s) {
                const int lc = s * 4 + 2 * half;        // local channel pair
                v2f a0, a1, b0, b1;
                a0.x = zT[0][c0b + lc][nl];     a0.y = zT[0][c0b + lc + 1][nl];
                a1.x = zT[1][c0b + lc][nl];     a1.y = zT[1][c0b + lc + 1][nl];
                b0.x = bB[buf][wave * 32 + nl][lc];      b0.y = bB[buf][wave * 32 + nl][lc + 1];
                b1.x = bB[buf][wave * 32 + 16 + nl][lc]; b1.y = bB[buf][wave * 32 + 16 + nl][lc + 1];
                acc[0][0] = __builtin_amdgcn_wmma_f32_16x16x4_f32(false, a0, false, b0, (short)0, acc[0][0], false, false);
                acc[0][1] = __builtin_amdgcn_wmma_f32_16x16x4_f32(false, a0, false, b1, (short)0, acc[0][1], false, false);
                acc[1][0] = __builtin_amdgcn_wmma_f32_16x16x4_f32(false, a1, false, b0, (short)0, acc[1][0], false, false);
                acc[1][1] = __builtin_amdgcn_wmma_f32_16x16x4_f32(false, a1, false, b1, (short)0, acc[1][1], false, false);
            }
            __syncthreads();                    // chunk `buf` free for restaging
            buf ^= 1;
        }

        // score = ||e||^2 - 2*dot ; elem r of lane -> (m = r+8*half, code n = nl)
        #pragma unroll
        for (int t = 0; t < 2; ++t) {
            const int kcode = kbase + wave * 32 + t * 16 + nl;
            const float en  = enorm[kcode];
            #pragma unroll
            for (int mt = 0; mt < 2; ++mt)
                #pragma unroll
                for (int r = 0; r < 8; ++r) {
                    float sc = en - 2.0f * acc[mt][t][r];
                    if (sc < best[mt][r]) { best[mt][r] = sc; bidx[mt][r] = kcode; }
                }
        }
    }

    // ---- reduce across the 16-lane N group (ties -> lowest index) ----
    #pragma unroll
    for (int mt = 0; mt < 2; ++mt)
        #pragma unroll
        for (int r = 0; r < 8; ++r) {
            float v  = best[mt][r];
            int   id = bidx[mt][r];
            #pragma unroll
            for (int mask = 8; mask >= 1; mask >>= 1) {
                float ov  = __shfl_xor(v,  mask, 32);
                int   oid = __shfl_xor(id, mask, 32);
                if (ov < v || (ov == v && oid < id)) { v = ov; id = oid; }
            }
            if (nl == 0) {
                wBest[wave][mt * 16 + r + 8 * half] = v;
                wIdx[wave][mt * 16 + r + 8 * half]  = id;
            }
        }
    __syncthreads();

    // ---- reduce across the 8 waves (disjoint K ranges) ----
    if (tid < MROWS) {
        float v  = wBest[0][tid];
        int   id = wIdx[0][tid];
        #pragma unroll
        for (int w2 = 1; w2 < 8; ++w2) {
            float ov  = wBest[w2][tid];
            int   oid = wIdx[w2][tid];
            if (ov < v || (ov == v && oid < id)) { v = ov; id = oid; }
        }
        finalIdx[tid] = id;
        idx_out[m0 + tid] = (float)id;
    }
    __syncthreads();

    // ---- write z_out = z + (zq - z) (reference rounding) + loss partial ----
    float ls = 0.f;
    #pragma unroll
    for (int j = 0; j < MROWS; ++j) {
        int li = j * 256 + tid;
        int c  = li >> 5;
        int mi = li & 31;
        float zv = zT[mi >> 4][c][mi & 15];
        float q  = e[(size_t)finalIdx[mi] * CDIM + c];
        float d  = q - zv;
        ls += d * d;
        z_out[zbase + (long long)c * HWSZ + mi] = zv + d;
    }
    #pragma unroll
    for (int mask = 16; mask >= 1; mask >>= 1) ls += __shfl_xor(ls, mask, 32);
    if (lane == 0) lred[wave] = ls;
    __syncthreads();
    if (tid == 0) {
        float s = 0.f;
        #pragma unroll
        for (int w2 = 0; w2 < 8; ++w2) s += lred[w2];
        loss_partial[blockIdx.x] = s;
    }
}

// ---------------------------------------------------------------------------
// Kernel 3: deterministic final loss reduction (fixed order, no atomics)
// ---------------------------------------------------------------------------
__global__ __launch_bounds__(256) void vq_loss(const float* __restrict__ partial,
                                               float* __restrict__ loss_out) {
    __shared__ float red[256];
    float s = 0.f;
    for (int i = threadIdx.x; i < NBLOCKS; i += 256) s += partial[i];
    red[threadIdx.x] = s;
    __syncthreads();
    for (int off = 128; off >= 1; off >>= 1) {
        if (threadIdx.x < off) red[threadIdx.x] += red[threadIdx.x + off];
        __syncthreads();
    }
    if (threadIdx.x == 0)
        loss_out[0] = red[0] / (float)(M_TOTAL * CDIM);
}

// ---------------------------------------------------------------------------
extern "C" void kernel_launch(void* const* d_in, const int* in_sizes, int n_in,
                              void* d_out, int out_size, void* d_ws, size_t ws_size,
                              hipStream_t stream) {
    const float* z = (const float*)d_in[0];   // [16,256,32,32]
    const float* e = (const float*)d_in[1];   // [8192,256]

    float* out_z    = (float*)d_out;                  // 4,194,304 floats
    float* out_loss = out_z + (size_t)M_TOTAL * CDIM; // 1 float
    float* out_idx  = out_loss + 1;                   // 16,384 floats

    float* enorm   = (float*)d_ws;                    // 8192 floats
    float* partial = enorm + K_CODES;                 // 512 floats

    vq_enorm<<<K_CODES * 32 / 256, 256, 0, stream>>>(e, enorm);
    vq_main <<<NBLOCKS, 256, 0, stream>>>(z, e, enorm, out_z, out_idx, partial);
    vq_loss <<<1, 256, 0, stream>>>(partial, out_loss);
}